// NeuralNetwork_32710470926988
// MI455X (gfx1250) — compile-verified
//
#include <hip/hip_runtime.h>
#include <hip/hip_bf16.h>

typedef __attribute__((ext_vector_type(2))) float v2f;
typedef __attribute__((ext_vector_type(8))) float v8f;

#define BROWS   8192
#define NPTS    4096
#define HID     64
#define ALPHA_F 0.3f
#define DECAY_F 0.7f
#define CHUNK   512       // 16 chunks of 512 rows; one wave per (chunk, 16-col strip)
#define NCHUNK  16
#define FIXSUB  4         // fix first 4*64 = 256 rows of each chunk (f^257 ~ 1e-40)

// f^1 .. f^8 and f^8 for the in-register scan
__constant__ const float FPOW[9] = {
    1.0f, 0.7f, 0.49f, 0.343f, 0.2401f, 0.16807f, 0.117649f, 0.0823543f, 0.05764801f
};

// ---------------------------------------------------------------------------
// Kernel 1: h = relu(relu(x*W1+b1) @ W2 + b2)   [8192 x 64] -> workspace
// ---------------------------------------------------------------------------
__global__ void mlp_h_kernel(const float* __restrict__ x,
                             const float* __restrict__ W1,
                             const float* __restrict__ b1,
                             const float* __restrict__ W2,
                             const float* __restrict__ b2,
                             float* __restrict__ h) {
    __shared__ float sh1[4][HID];
    const int j   = threadIdx.x & 63;
    const int r   = threadIdx.x >> 6;          // 0..3
    const int row = blockIdx.x * 4 + r;

    const float xv = x[row];                   // x is [B,1]
    float v = fmaf(xv, W1[j], b1[j]);
    sh1[r][j] = fmaxf(v, 0.0f);
    __syncthreads();

    float acc = b2[j];
#pragma unroll 8
    for (int k = 0; k < HID; ++k)
        acc = fmaf(sh1[r][k], W2[k * HID + j], acc);
    h[row * HID + j] = fmaxf(acc, 0.0f);
}

// ---------------------------------------------------------------------------
// Kernel 2 (fused): y-tile GEMM (V_WMMA_F32_16X16X4_F32) + in-register EWMA
// local scan. One wave per (chunk, col-strip): 32 M-tiles of 16x16 walked
// top-to-bottom; W3 fragment register-resident across all 32 tiles.
// C layout: lane l (p = l&15, half = l>>4), VGPR v holds (row v+8*half, col p).
// ---------------------------------------------------------------------------
__global__ void gemm_ewma_kernel(const float* __restrict__ h,
                                 const float* __restrict__ W3,
                                 const float* __restrict__ b3,
                                 float* __restrict__ y,
                                 float* __restrict__ carry) {
    const int wave  = blockIdx.x * 8 + (threadIdx.x >> 5);  // 4096 waves
    const int lane  = threadIdx.x & 31;
    const int chunk = wave >> 8;                            // 0..15
    const int nt    = wave & 255;                           // col-strip 0..255
    const int n0    = nt << 4;
    const int half  = lane >> 4;
    const int p     = lane & 15;

    // ---- preload B fragments: W3[0:64, n0:n0+16], 16 K-steps of 4 ----------
    v2f bfr[16];
    {
        const float* bp = W3 + (size_t)(half * 2) * NPTS + (n0 + p);
#pragma unroll
        for (int kk = 0; kk < 16; ++kk) {
            bfr[kk].x = bp[0];
            bfr[kk].y = bp[NPTS];
            bp += (size_t)4 * NPTS;
        }
    }

    const float bias = b3[n0 + p];
    const float F8   = FPOW[8];
    float R = 0.0f;                        // running carry (uniform per column)
    const int rowBase = chunk * CHUNK;

    for (int tile = 0; tile < 32; ++tile) {
        const int m0 = rowBase + tile * 16;

        // ---- GEMM: 16 x WMMA f32 16x16x4 -------------------------------
        v8f c = {bias, bias, bias, bias, bias, bias, bias, bias};
        const float* ap = h + (size_t)(m0 + p) * HID + half * 2;
#pragma unroll
        for (int kk = 0; kk < 16; ++kk) {
            v2f a;
            a.x = ap[0];
            a.y = ap[1];
            c = __builtin_amdgcn_wmma_f32_16x16x4_f32(
                    false, a, false, bfr[kk], (short)0, c, false, false);
            ap += 4;
        }

        // ---- in-register EWMA over the 16 rows of this tile ------------
        // zero-carry local scan of this half's 8 rows
        const bool firstrow = (chunk == 0) && (tile == 0) && (half == 0);
        float l[8];
        {
            float v0 = ALPHA_F * c[0];
            if (firstrow) v0 = c[0];               // r[0] = y[0]
            l[0] = v0;
            float prev = v0;
#pragma unroll
            for (int t = 1; t < 8; ++t) {
                prev = fmaf(DECAY_F, prev, ALPHA_F * c[t]);
                l[t] = prev;
            }
        }
        // broadcast each half's local final (l[7]) to both halves
        const int myl7 = __float_as_int(l[7]);
        const float L0 = __int_as_float(__builtin_amdgcn_ds_bpermute(p * 4, myl7));
        const float L1 = __int_as_float(__builtin_amdgcn_ds_bpermute((p + 16) * 4, myl7));
        // carry into half1 = half0 final with incoming carry applied
        const float C1  = fmaf(F8, R, L0);
        const float cin = half ? C1 : R;
        // apply carry correction: c[t] = l[t] + f^(t+1) * cin
#pragma unroll
        for (int t = 0; t < 8; ++t)
            c[t] = fmaf(FPOW[t + 1], cin, l[t]);
        // carry out of this tile = half1 final with its carry applied
        R = fmaf(F8, C1, L1);

        // ---- store the scanned tile ------------------------------------
        float* outp = y + (size_t)(m0 + half * 8) * NPTS + n0 + p;
#pragma unroll
        for (int v = 0; v < 8; ++v)
            outp[(size_t)v * NPTS] = c[v];
    }

    // chunk's final value (exact global carry: f^512 == 0 in fp32)
    if (lane < 16)
        carry[chunk * NPTS + n0 + p] = R;
}

// ---------------------------------------------------------------------------
// Kernel 3: cross-chunk carry fixup. r_true[c*512+t] = l[t] + f^(t+1)*carry[c-1].
// Only the first 256 rows of each chunk need it (f^257 ~ 1e-40).
// ---------------------------------------------------------------------------
__global__ void ewma_fixup_kernel(float* __restrict__ y,
                                  const float* __restrict__ carry) {
    const int tid  = blockIdx.x * blockDim.x + threadIdx.x;    // 15*4096*4
    const int col  = tid & (NPTS - 1);
    const int rest = tid >> 12;                                // 0..59
    const int chunk = 1 + (rest % (NCHUNK - 1));               // 1..15
    const int sub   = rest / (NCHUNK - 1);                     // 0..3

    const float L = carry[(chunk - 1) * NPTS + col];
    const int t0 = sub * 64;

    float fac = DECAY_F;                                       // f^(t0+1)
    for (int i = 0; i < t0; ++i) fac *= DECAY_F;

    size_t base = ((size_t)chunk * CHUNK + t0) * NPTS + col;
#pragma unroll 8
    for (int t = 0; t < 64; ++t) {
        y[base + (size_t)t * NPTS] += fac * L;
        fac *= DECAY_F;
    }
}

// ---------------------------------------------------------------------------
extern "C" void kernel_launch(void* const* d_in, const int* in_sizes, int n_in,
                              void* d_out, int out_size, void* d_ws, size_t ws_size,
                              hipStream_t stream) {
    const float* x  = (const float*)d_in[0];   // [8192, 1]
    const float* W1 = (const float*)d_in[1];   // [1, 64]
    const float* b1 = (const float*)d_in[2];   // [64]
    const float* W2 = (const float*)d_in[3];   // [64, 64]
    const float* b2 = (const float*)d_in[4];   // [64]
    const float* W3 = (const float*)d_in[5];   // [64, 4096]
    const float* b3 = (const float*)d_in[6];   // [4096]
    float* out = (float*)d_out;                // [8192, 4096]

    float* h     = (float*)d_ws;                                              // 2 MiB
    float* carry = (float*)((char*)d_ws + (size_t)BROWS * HID * sizeof(float)); // 256 KiB

    // 1) h = relu(relu(x W1 + b1) W2 + b2)
    mlp_h_kernel<<<BROWS / 4, 256, 0, stream>>>(x, W1, b1, W2, b2, h);

    // 2) fused GEMM (fp32 WMMA) + local EWMA scan, straight to d_out
    //    16 chunks * 256 col-strips = 4096 waves; 8 waves/block -> 512 blocks
    gemm_ewma_kernel<<<512, 256, 0, stream>>>(h, W3, b3, out, carry);

    // 3) cross-chunk carry fixup
    ewma_fixup_kernel<<<((NCHUNK - 1) * NPTS * FIXSUB) / 256, 256, 0, stream>>>(out, carry);
}